// KalmanFilterUpdate_16509854286246
// MI455X (gfx1250) — compile-verified
//
#include <hip/hip_runtime.h>
#include <stdint.h>

// MI455X / gfx1250, wave32. Batched Kalman update, B=262144, DX=8, DZ=4.
// Memory-bound target: ~151 MB traffic -> ~6.5 us @ 23.3 TB/s.

typedef __attribute__((ext_vector_type(2))) float v2f;
typedef __attribute__((ext_vector_type(8))) float v8f;

#define BPB 128          // batches per block (= threads per block)
#define PSTRIDE 68       // LDS floats per batch for P: 272B, 16B aligned, low bank conflict
#define PHTSTRIDE 36     // LDS floats per batch for PHT (8x4 + pad)

__global__ __launch_bounds__(128) void kalman_kernel(
    const float* __restrict__ x, const float* __restrict__ z,
    const float* __restrict__ P, const float* __restrict__ H,
    const float* __restrict__ R, float* __restrict__ out)
{
  __shared__ float p_sh[BPB * PSTRIDE];      // 34816 B
  __shared__ float pht_sh[BPB * PHTSTRIDE];  // 18432 B

  const int t = threadIdx.x;
  const int blockBase = blockIdx.x * BPB;

  // ---------------- Stage P: global -> LDS, CDNA5 async copy, coalesced B128 ----------------
  {
    const char* gp = (const char*)(P + (size_t)blockBase * 64);
    const unsigned ldsBase = (unsigned)(uintptr_t)(void*)p_sh;  // AS3 offset in low 32 bits
#pragma unroll
    for (int j = 0; j < 16; ++j) {
      const int c = j * BPB + t;                  // chunk id: lanes -> consecutive 16B in memory
      const char* ga = gp + (size_t)c * 16;
      const unsigned lo = ldsBase + (unsigned)((c >> 4) * (PSTRIDE * 4) + (c & 15) * 16);
      asm volatile("global_load_async_to_lds_b128 %0, %1, off"
                   :: "v"(lo), "v"(ga) : "memory");
    }
    asm volatile("s_wait_asynccnt 0" ::: "memory");
  }
  __syncthreads();

  // ---------------- PHT = P * H^T via V_WMMA_F32_16X16X4_F32 (2 batches / issue) -------------
  // A (16x8): rows 0..7 = batch even, rows 8..15 = batch odd.  B = H^T (8x4), cols 4..15 zero.
  const int lane = t & 31;
  const int wave = t >> 5;
  const int rowM = lane & 15;         // A row / C column
  const int half = lane >> 4;         // K sub-block select per ISA layout
  const int n    = rowM;              // C column this lane holds (valid if < 4)
  const int kb   = 2 * half;

  v2f b0 = {0.f, 0.f}, b1 = {0.f, 0.f};
  if (n < 4) {                        // B layout: lane=N, VGPR v -> K = v + 2*half (+4 per step)
    b0.x = H[n * 8 + kb + 0]; b0.y = H[n * 8 + kb + 1];
    b1.x = H[n * 8 + 4 + kb + 0]; b1.y = H[n * 8 + 4 + kb + 1];
  }

#pragma unroll
  for (int p = 0; p < 16; ++p) {
    const int pairBase = wave * 32 + 2 * p;            // block-local batch pair (same wave)
    const int bl = pairBase + (rowM >> 3);
    const int r  = rowM & 7;
    const float* ap = &p_sh[bl * PSTRIDE + r * 8 + kb]; // 8B-aligned
    v2f a0 = *(const v2f*)(ap);          // K = kb, kb+1
    v2f a1 = *(const v2f*)(ap + 4);      // K = 4+kb, 4+kb+1
    v8f c = {0.f, 0.f, 0.f, 0.f, 0.f, 0.f, 0.f, 0.f};
    c = __builtin_amdgcn_wmma_f32_16x16x4_f32(false, a0, false, b0, (short)0, c, false, false);
    c = __builtin_amdgcn_wmma_f32_16x16x4_f32(false, a1, false, b1, (short)0, c, false, false);
    if (n < 4) {                          // C: VGPR r2 -> row r2, lane-half -> which batch
      const int ob = (pairBase + half) * PHTSTRIDE + n;
#pragma unroll
      for (int r2 = 0; r2 < 8; ++r2) pht_sh[ob + r2 * 4] = c[r2];
    }
  }
  __syncthreads();

  // ---------------- Per-lane fp32 phase: one batch per thread ----------------
  const size_t b = (size_t)blockBase + t;

  float xv[8];
#pragma unroll
  for (int i = 0; i < 8; ++i) xv[i] = x[b * 8 + i];
  float zv[4];
#pragma unroll
  for (int i = 0; i < 4; ++i) zv[i] = z[b * 4 + i];

  float pht[32];
#pragma unroll
  for (int i = 0; i < 32; ++i) pht[i] = pht_sh[t * PHTSTRIDE + i];

  // S = H * PHT + R   (H, R uniform -> scalar loads)
  float S[16];
#pragma unroll
  for (int m = 0; m < 4; ++m)
#pragma unroll
    for (int nn = 0; nn < 4; ++nn) {
      float acc = R[m * 4 + nn];
#pragma unroll
      for (int k = 0; k < 8; ++k) acc += H[m * 8 + k] * pht[k * 4 + nn];
      S[m * 4 + nn] = acc;
    }

  // SI = inv(S), 4x4 adjugate
  float SI[16];
  {
    const float a00=S[0],a01=S[1],a02=S[2],a03=S[3];
    const float a10=S[4],a11=S[5],a12=S[6],a13=S[7];
    const float a20=S[8],a21=S[9],a22=S[10],a23=S[11];
    const float a30=S[12],a31=S[13],a32=S[14],a33=S[15];
    const float s0=a00*a11-a01*a10, s1=a00*a12-a02*a10, s2=a00*a13-a03*a10;
    const float s3=a01*a12-a02*a11, s4=a01*a13-a03*a11, s5=a02*a13-a03*a12;
    const float c5=a22*a33-a23*a32, c4=a21*a33-a23*a31, c3=a21*a32-a22*a31;
    const float c2=a20*a33-a23*a30, c1=a20*a32-a22*a30, c0=a20*a31-a21*a30;
    const float det = s0*c5 - s1*c4 + s2*c3 + s3*c2 - s4*c1 + s5*c0;
    const float id = 1.0f / det;
    SI[0]  = ( a11*c5 - a12*c4 + a13*c3) * id;
    SI[1]  = (-a01*c5 + a02*c4 - a03*c3) * id;
    SI[2]  = ( a31*s5 - a32*s4 + a33*s3) * id;
    SI[3]  = (-a21*s5 + a22*s4 - a23*s3) * id;
    SI[4]  = (-a10*c5 + a12*c2 - a13*c1) * id;
    SI[5]  = ( a00*c5 - a02*c2 + a03*c1) * id;
    SI[6]  = (-a30*s5 + a32*s2 - a33*s1) * id;
    SI[7]  = ( a20*s5 - a22*s2 + a23*s1) * id;
    SI[8]  = ( a10*c4 - a11*c2 + a13*c0) * id;
    SI[9]  = (-a00*c4 + a01*c2 - a03*c0) * id;
    SI[10] = ( a30*s4 - a31*s2 + a33*s0) * id;
    SI[11] = (-a20*s4 + a21*s2 - a23*s0) * id;
    SI[12] = (-a10*c3 + a11*c1 - a12*c0) * id;
    SI[13] = ( a00*c3 - a01*c1 + a02*c0) * id;
    SI[14] = (-a30*s3 + a31*s1 - a32*s0) * id;
    SI[15] = ( a20*s3 - a21*s1 + a22*s0) * id;
  }

  // K = PHT * SI
  float K[32];
#pragma unroll
  for (int i = 0; i < 8; ++i)
#pragma unroll
    for (int nn = 0; nn < 4; ++nn) {
      float acc = 0.f;
#pragma unroll
      for (int m = 0; m < 4; ++m) acc += pht[i * 4 + m] * SI[m * 4 + nn];
      K[i * 4 + nn] = acc;
    }

  // y = z - H x;  x_new = x + K y
  float yv[4];
#pragma unroll
  for (int m = 0; m < 4; ++m) {
    float acc = zv[m];
#pragma unroll
    for (int k = 0; k < 8; ++k) acc -= H[m * 8 + k] * xv[k];
    yv[m] = acc;
  }
  float res[72];
#pragma unroll
  for (int i = 0; i < 8; ++i) {
    float acc = xv[i];
#pragma unroll
    for (int m = 0; m < 4; ++m) acc += K[i * 4 + m] * yv[m];
    res[i] = acc;
  }

  // IKH = I - K H
  float ikh[64];
#pragma unroll
  for (int i = 0; i < 8; ++i)
#pragma unroll
    for (int j = 0; j < 8; ++j) {
      float acc = (i == j) ? 1.0f : 0.0f;
#pragma unroll
      for (int m = 0; m < 4; ++m) acc -= K[i * 4 + m] * H[m * 8 + j];
      ikh[i * 8 + j] = acc;
    }

  // W = P * IKH^T  (stream P from LDS: each element reused 8x)
  float W[64];
#pragma unroll
  for (int i = 0; i < 64; ++i) W[i] = 0.f;
#pragma unroll
  for (int k = 0; k < 8; ++k)
#pragma unroll
    for (int m = 0; m < 8; ++m) {
      const float pv = p_sh[t * PSTRIDE + k * 8 + m];
#pragma unroll
      for (int j = 0; j < 8; ++j) W[k * 8 + j] += pv * ikh[j * 8 + m];
    }

  // KR = K * R
  float KR[32];
#pragma unroll
  for (int i = 0; i < 8; ++i)
#pragma unroll
    for (int nn = 0; nn < 4; ++nn) {
      float acc = 0.f;
#pragma unroll
      for (int m = 0; m < 4; ++m) acc += K[i * 4 + m] * R[m * 4 + nn];
      KR[i * 4 + nn] = acc;
    }

  // P_new = IKH * W + KR * K^T  (symmetric: compute upper triangle, mirror)
#pragma unroll
  for (int i = 0; i < 8; ++i)
#pragma unroll
    for (int j = 0; j < 8; ++j) {
      if (j < i) continue;
      float acc = 0.f;
#pragma unroll
      for (int m = 0; m < 8; ++m) acc += ikh[i * 8 + m] * W[m * 8 + j];
#pragma unroll
      for (int nn = 0; nn < 4; ++nn) acc += KR[i * 4 + nn] * K[j * 4 + nn];
      res[8 + i * 8 + j] = acc;
      res[8 + j * 8 + i] = acc;
    }

  // ---------------- store: 18 aligned float4 per batch ----------------
  float4* op = reinterpret_cast<float4*>(out + b * 72);
#pragma unroll
  for (int q = 0; q < 18; ++q)
    op[q] = make_float4(res[4 * q + 0], res[4 * q + 1], res[4 * q + 2], res[4 * q + 3]);
}

extern "C" void kernel_launch(void* const* d_in, const int* in_sizes, int n_in,
                              void* d_out, int out_size, void* d_ws, size_t ws_size,
                              hipStream_t stream) {
  (void)in_sizes; (void)n_in; (void)d_ws; (void)ws_size; (void)out_size;
  const float* x = (const float*)d_in[0];
  const float* z = (const float*)d_in[1];
  const float* P = (const float*)d_in[2];
  const float* H = (const float*)d_in[3];
  const float* R = (const float*)d_in[4];
  float* out = (float*)d_out;
  const int B = 262144;
  dim3 grid(B / BPB), block(BPB);
  hipLaunchKernelGGL(kalman_kernel, grid, block, 0, stream, x, z, P, H, R, out);
}